// GCNModelWithFocalLoss_6090263626384
// MI455X (gfx1250) — compile-verified
//
#include <hip/hip_runtime.h>

typedef __attribute__((ext_vector_type(2))) float v2f;
typedef __attribute__((ext_vector_type(8))) float v8f;

#define D_IN  128
#define D_HID 128
#define D_OUT 64

// ---------------- degree / normalization ----------------

__global__ void init_deg_kernel(float* deg, int n) {
    int i = blockIdx.x * blockDim.x + threadIdx.x;
    if (i < n) deg[i] = 1.0f;               // self-loop contributes 1
}

__global__ void accum_deg_kernel(const int* __restrict__ dst, float* deg, int E) {
    int e = blockIdx.x * blockDim.x + threadIdx.x;
    if (e < E) atomicAdd(&deg[dst[e]], 1.0f);
}

__global__ void dinv_kernel(float* deg, int n) {
    int i = blockIdx.x * blockDim.x + threadIdx.x;
    if (i < n) deg[i] = rsqrtf(deg[i]);     // deg >= 1 always (self-loop)
}

__global__ void zero_kernel(float* p, long long n) {
    long long i = (long long)blockIdx.x * blockDim.x + threadIdx.x;
    if (i < n) p[i] = 0.0f;
}

// ---------------- fp32 WMMA GEMM: C[M,Nc] = A[M,K] * W[K,Nc] ----------------
// One wave (32 lanes) computes one 16x16 output tile, K-loop in steps of 4
// using V_WMMA_F32_16X16X4_F32. Fragment layouts per CDNA5 ISA 7.12.2:
//   A 16x4 : lanes 0-15 hold K=0,1 in v[0],v[1]; lanes 16-31 hold K=2,3
//   B 4x16 : v[0] = row K=0 (lanes 0-15) / K=2 (lanes 16-31); v[1] = K=1 / K=3
//   C 16x16: v[0..7] = rows M=0..7 (lanes 0-15) / M=8..15 (lanes 16-31)
__global__ void gemm_wmma_f32(const float* __restrict__ A,
                              const float* __restrict__ W,
                              float* __restrict__ C,
                              int M, int K, int Nc) {
    __shared__ float lds_a[16 * 128];           // 16 rows x K (K<=128) = 8 KB
    const int m0   = blockIdx.x * 16;
    const int n0   = blockIdx.y * 16;
    const int lane = threadIdx.x;               // wave32: 0..31
    if (m0 >= M) return;

    // Coalesced float4 stage of the 16xK A-tile into LDS.
    const int chunks_per_row = K / 4;
    const int nchunks = 16 * chunks_per_row;
    for (int idx = lane; idx < nchunks; idx += 32) {
        int r = idx / chunks_per_row;
        int c = idx % chunks_per_row;
        ((float4*)lds_a)[idx] =
            ((const float4*)(A + (size_t)(m0 + r) * K))[c];
    }
    __syncthreads();

    const int halfsel = lane >> 4;              // 0: lanes 0-15, 1: lanes 16-31
    const int rlo     = lane & 15;
    const int kh      = halfsel * 2;            // K sub-offset owned by this half
    const int col     = n0 + rlo;

    v8f acc = {};
    for (int k0 = 0; k0 < K; k0 += 4) {
        v2f a, b;
        a.x = lds_a[rlo * K + k0 + kh + 0];
        a.y = lds_a[rlo * K + k0 + kh + 1];
        b.x = W[(size_t)(k0 + kh + 0) * Nc + col];
        b.y = W[(size_t)(k0 + kh + 1) * Nc + col];
        // 8 args: (neg_a, A, neg_b, B, c_mod, C, reuse_a, reuse_b)
        acc = __builtin_amdgcn_wmma_f32_16x16x4_f32(
            false, a, false, b, (short)0, acc, false, false);
    }

    const int rbase = m0 + halfsel * 8;
    #pragma unroll
    for (int v = 0; v < 8; ++v)
        C[(size_t)(rbase + v) * Nc + col] = acc[v];
}

// ---------------- edge-parallel gather + scale + scatter-add ----------------
template <int D>
__global__ void aggregate_kernel(const float* __restrict__ H,
                                 const int* __restrict__ src,
                                 const int* __restrict__ dst,
                                 const float* __restrict__ dinv,
                                 float* __restrict__ out, int E) {
    constexpr int EPB = 256 / D;                // edges per 256-thread block
    int e = blockIdx.x * EPB + (int)(threadIdx.x / D);
    int f = threadIdx.x % D;
    if (e < E) {
        int s = src[e], d = dst[e];
        float norm = dinv[s] * dinv[d];
        atomicAdd(&out[(size_t)d * D + f], H[(size_t)s * D + f] * norm);
    }
}

// self-loop term + bias + ReLU (layer 1)
__global__ void finish1_kernel(float* __restrict__ A1,
                               const float* __restrict__ H1,
                               const float* __restrict__ dinv,
                               const float* __restrict__ b, int n) {
    int i = blockIdx.x * blockDim.x + threadIdx.x;
    if (i < n * D_HID) {
        int node = i / D_HID, f = i % D_HID;
        float di = dinv[node];
        float v  = A1[i] + H1[i] * di * di + b[f];
        A1[i] = v > 0.0f ? v : 0.0f;
    }
}

// self-loop term + bias + log_softmax over 64 cols; one wave32 per node,
// 2 columns per lane, cross-lane reductions via __shfl_xor (width 32).
__global__ void finish2_lsm_kernel(float* __restrict__ out,
                                   const float* __restrict__ H2,
                                   const float* __restrict__ dinv,
                                   const float* __restrict__ b, int n) {
    int node = blockIdx.x * (blockDim.x / 32) + (int)(threadIdx.x / 32);
    int lane = threadIdx.x & 31;
    if (node >= n) return;
    float di = dinv[node];
    float d2 = di * di;
    size_t base = (size_t)node * D_OUT;
    float v0 = out[base + lane]      + H2[base + lane]      * d2 + b[lane];
    float v1 = out[base + lane + 32] + H2[base + lane + 32] * d2 + b[lane + 32];

    float m = fmaxf(v0, v1);
    #pragma unroll
    for (int off = 16; off > 0; off >>= 1)
        m = fmaxf(m, __shfl_xor(m, off, 32));
    float s = __expf(v0 - m) + __expf(v1 - m);
    #pragma unroll
    for (int off = 16; off > 0; off >>= 1)
        s += __shfl_xor(s, off, 32);
    float lse = m + __logf(s);

    out[base + lane]      = v0 - lse;
    out[base + lane + 32] = v1 - lse;
}

// ---------------- launch ----------------

extern "C" void kernel_launch(void* const* d_in, const int* in_sizes, int n_in,
                              void* d_out, int out_size, void* d_ws, size_t ws_size,
                              hipStream_t stream) {
    const float* x  = (const float*)d_in[0];
    const int*   ei = (const int*)  d_in[1];   // [2, E] int32
    const float* W1 = (const float*)d_in[2];
    const float* b1 = (const float*)d_in[3];
    const float* W2 = (const float*)d_in[4];
    const float* b2 = (const float*)d_in[5];

    const int E = in_sizes[1] / 2;
    const int N = in_sizes[0] / D_IN;          // 50000 (divisible by 16)
    const int* src = ei;
    const int* dst = ei + E;

    float* ws   = (float*)d_ws;
    float* dinv = ws;                          // N
    float* H1   = dinv + N;                    // N * 128
    float* A1   = H1 + (size_t)N * D_HID;      // N * 128
    float* H2   = A1 + (size_t)N * D_HID;      // N * 64
    float* outf = (float*)d_out;               // N * 64 (also layer-2 accumulator)

    // normalization
    init_deg_kernel <<<(N + 255) / 256, 256, 0, stream>>>(dinv, N);
    accum_deg_kernel<<<(E + 255) / 256, 256, 0, stream>>>(dst, dinv, E);
    dinv_kernel     <<<(N + 255) / 256, 256, 0, stream>>>(dinv, N);

    // layer 1: H1 = X * W1  (WMMA), aggregate, self-loop + bias + ReLU
    {
        dim3 grid(N / 16, D_HID / 16);
        gemm_wmma_f32<<<grid, 32, 0, stream>>>(x, W1, H1, N, D_IN, D_HID);
    }
    {
        long long n1 = (long long)N * D_HID;
        zero_kernel<<<(unsigned)((n1 + 255) / 256), 256, 0, stream>>>(A1, n1);
        aggregate_kernel<D_HID><<<(E + 1) / 2, 256, 0, stream>>>(H1, src, dst, dinv, A1, E);
        finish1_kernel<<<(unsigned)((n1 + 255) / 256), 256, 0, stream>>>(A1, H1, dinv, b1, N);
    }

    // layer 2: H2 = A1 * W2 (WMMA), aggregate into d_out, finish + log_softmax
    {
        dim3 grid(N / 16, D_OUT / 16);
        gemm_wmma_f32<<<grid, 32, 0, stream>>>(A1, W2, H2, N, D_HID, D_OUT);
    }
    {
        long long n2 = (long long)N * D_OUT;
        zero_kernel<<<(unsigned)((n2 + 255) / 256), 256, 0, stream>>>(outf, n2);
        aggregate_kernel<D_OUT><<<(E + 3) / 4, 256, 0, stream>>>(H2, src, dst, dinv, outf, E);
        finish2_lsm_kernel<<<(N + 7) / 8, 256, 0, stream>>>(outf, H2, dinv, b2, N);
    }
}